// ModelNew_5909874999867
// MI455X (gfx1250) — compile-verified
//
#include <hip/hip_runtime.h>

typedef float v2f __attribute__((ext_vector_type(2)));
typedef float v8f __attribute__((ext_vector_type(8)));

#define NN 8192
#define BM 128
#define BN 128
#define BK 32
#define TGRID (NN / BM)   // 64 tiles per dim
#define LDA 36            // padded floats per A row in LDS (16B aligned, conflict-free)
#define LDB 136           // padded floats per B row in LDS (16B aligned, conflict-free)

__global__ __launch_bounds__(256) void tril_wmma_f32_gemm(
    const float* __restrict__ A, const float* __restrict__ B,
    float* __restrict__ C) {
  __shared__ __align__(16) float As[BM * LDA];   // 128 x 36 floats = 18 KB
  __shared__ __align__(16) float Bs[BK * LDB];   // 32 x 136 floats = 17 KB

  // Map linear block id -> lower-triangular tile (bm >= bn).
  const int id = blockIdx.x;
  int bm = (int)((sqrtf(8.0f * (float)id + 1.0f) - 1.0f) * 0.5f);
  while ((bm + 1) * (bm + 2) / 2 <= id) ++bm;
  while (bm * (bm + 1) / 2 > id) --bm;
  const int bn = id - bm * (bm + 1) / 2;

  const int tid  = threadIdx.x;
  const int wave = tid >> 5;
  const int lane = tid & 31;
  const int half = lane >> 4;   // selects K pair (A/B frags) and M+8 (C frag)
  const int l16  = lane & 15;
  const int wm   = wave >> 2;   // 0..1 -> 64 rows of the 128-row block
  const int wn   = wave & 3;    // 0..3 -> 32 cols of the 128-col block

  const v8f vzero = {0.f, 0.f, 0.f, 0.f, 0.f, 0.f, 0.f, 0.f};
  v8f acc[4][2];
#pragma unroll
  for (int i = 0; i < 4; ++i)
#pragma unroll
    for (int j = 0; j < 2; ++j) acc[i][j] = vzero;

  const int rowC0 = bm * BM;
  const int colC0 = bn * BN;

  // k-tile range implied by triangular structure: k in [bn*128, bm*128+128)
  const int ktBeg = bn * (BM / BK);
  const int ktEnd = bm * (BM / BK) + (BM / BK);

  float4 pa[4], pb[4];

  auto load_stage = [&](int kt, float4* ra, float4* rb) {
#pragma unroll
    for (int i = 0; i < 4; ++i) {
      const int q = tid + 256 * i;
      const int rowA = q >> 3, ca = (q & 7) << 2;   // 128 rows x 8 float4
      ra[i] = *(const float4*)(A + (size_t)(rowC0 + rowA) * NN + kt * BK + ca);
      const int rowB = q >> 5, cb = (q & 31) << 2;  // 32 rows x 32 float4
      rb[i] = *(const float4*)(B + (size_t)(kt * BK + rowB) * NN + colC0 + cb);
    }
  };
  auto store_stage = [&](const float4* ra, const float4* rb) {
#pragma unroll
    for (int i = 0; i < 4; ++i) {
      const int q = tid + 256 * i;
      const int rowA = q >> 3, ca = (q & 7) << 2;
      *(float4*)(&As[rowA * LDA + ca]) = ra[i];
      const int rowB = q >> 5, cb = (q & 31) << 2;
      *(float4*)(&Bs[rowB * LDB + cb]) = rb[i];
    }
  };

  load_stage(ktBeg, pa, pb);
  store_stage(pa, pb);
  __syncthreads();

  for (int kt = ktBeg; kt < ktEnd; ++kt) {
    const bool more = (kt + 1) < ktEnd;
    if (more) load_stage(kt + 1, pa, pb);  // prefetch next stage into registers

#pragma unroll
    for (int kk = 0; kk < BK; kk += 4) {
      // A fragments: 16x4 f32, lane l16 = row M, half selects K base {0,2}
      v2f af[4];
#pragma unroll
      for (int i = 0; i < 4; ++i) {
        const int rowA = wm * 64 + i * 16 + l16;
        af[i] = *(const v2f*)(&As[rowA * LDA + kk + 2 * half]);
      }
      // B fragments: 4x16 f32, lane l16 = col N, half selects K row pair
      v2f bf[2];
#pragma unroll
      for (int j = 0; j < 2; ++j) {
        const int colB = wn * 32 + j * 16 + l16;
        v2f t;
        t.x = Bs[(kk + 2 * half) * LDB + colB];
        t.y = Bs[(kk + 2 * half + 1) * LDB + colB];
        bf[j] = t;
      }
#pragma unroll
      for (int i = 0; i < 4; ++i)
#pragma unroll
        for (int j = 0; j < 2; ++j)
          acc[i][j] = __builtin_amdgcn_wmma_f32_16x16x4_f32(
              false, af[i], false, bf[j], (short)0, acc[i][j], false, false);
    }

    __syncthreads();
    if (more) {
      store_stage(pa, pb);
      __syncthreads();
    }
  }

  // C/D fragment layout: VGPR r -> M = r + 8*half, N = l16
#pragma unroll
  for (int i = 0; i < 4; ++i) {
#pragma unroll
    for (int j = 0; j < 2; ++j) {
      const int row0 = rowC0 + wm * 64 + i * 16 + 8 * half;
      const int col  = colC0 + wn * 32 + j * 16 + l16;
#pragma unroll
      for (int r = 0; r < 8; ++r)
        C[(size_t)(row0 + r) * NN + col] = acc[i][j][r];
    }
  }
}

// Zero-fill the whole output so the strictly-upper tiles (never touched by the
// triangular grid) are 0, matching tril(C). Lower tiles get overwritten.
__global__ __launch_bounds__(256) void zero_fill_f4(float4* __restrict__ p,
                                                    long n4) {
  long i = (long)blockIdx.x * 256 + threadIdx.x;
  const long stride = (long)gridDim.x * 256;
  const float4 z = make_float4(0.f, 0.f, 0.f, 0.f);
  for (; i < n4; i += stride) p[i] = z;
}

extern "C" void kernel_launch(void* const* d_in, const int* in_sizes, int n_in,
                              void* d_out, int out_size, void* d_ws,
                              size_t ws_size, hipStream_t stream) {
  (void)in_sizes; (void)n_in; (void)out_size; (void)d_ws; (void)ws_size;
  const float* A = (const float*)d_in[0];
  const float* B = (const float*)d_in[1];
  float* C = (float*)d_out;

  const long n4 = (long)NN * NN / 4;
  zero_fill_f4<<<4096, 256, 0, stream>>>((float4*)C, n4);

  const int nTiles = TGRID * (TGRID + 1) / 2;  // 2080 lower-tri tiles
  tril_wmma_f32_gemm<<<nTiles, 256, 0, stream>>>(A, B, C);
}